// GATModel_23880018165826
// MI455X (gfx1250) — compile-verified
//
#include <hip/hip_runtime.h>
#include <hip/hip_bf16.h>

#define DD   100   // hidden dim
#define HH   4     // heads
#define PH   100   // per-head dim
#define HD   400   // H*PH
#define LY   6     // layers
#define ODIM 64

typedef float v2f __attribute__((ext_vector_type(2)));
typedef float v8f __attribute__((ext_vector_type(8)));

__device__ __forceinline__ float wave_sum(float v) {
  #pragma unroll
  for (int off = 16; off; off >>= 1) v += __shfl_xor(v, off, 32);
  return v;
}

// ---------------- utility fills ----------------
__global__ void fill_f32_kernel(float* p, float v, long n) {
  long i = (long)blockIdx.x * blockDim.x + threadIdx.x;
  if (i < n) p[i] = v;
}
__global__ void fill_u32_kernel(unsigned int* p, unsigned int v, long n) {
  long i = (long)blockIdx.x * blockDim.x + threadIdx.x;
  if (i < n) p[i] = v;
}

// ---------------- h = emb[x] ----------------
__global__ void embed_kernel(const int* __restrict__ x, const float* __restrict__ emb,
                             float* __restrict__ h, int N) {
  long i = (long)blockIdx.x * blockDim.x + threadIdx.x;
  if (i < (long)N * DD) {
    int n = (int)(i / DD), d = (int)(i % DD);
    h[i] = emb[(long)x[n] * DD + d];
  }
}

// ---------------- hx[N,400] = h[N,100] @ W^T  (fp32 WMMA 16x16x4) ----------------
// One wave per 16-row strip of nodes. A tile (16x100) preloaded into 25 v2f regs,
// reused across all 25 output column tiles.
__global__ void gemm_hx_kernel(const float* __restrict__ h, const float* __restrict__ Wl,
                               float* __restrict__ hx, int N) {
  int wave = (int)(((long)blockIdx.x * blockDim.x + threadIdx.x) >> 5);
  int lane = threadIdx.x & 31;
  int ntiles = (N + 15) >> 4;
  if (wave >= ntiles) return;
  int l16 = lane & 15;
  int hi  = lane >> 4;          // 0: K pair {0,1}; 1: K pair {2,3}
  int row = wave * 16 + l16;
  int rowc = row < N ? row : N - 1;      // clamp (N is a multiple of 16 here anyway)
  const float* hrow = h + (long)rowc * DD;

  v2f areg[25];
  #pragma unroll
  for (int kk = 0; kk < 25; ++kk) {
    int k = kk * 4 + hi * 2;
    areg[kk].x = hrow[k];
    areg[kk].y = hrow[k + 1];
  }

  for (int ot = 0; ot < 25; ++ot) {
    int colb = ot * 16;
    const float* wrow = Wl + (long)(colb + l16) * DD;  // B[k][n] = W[n][k]
    v8f c = {0.f, 0.f, 0.f, 0.f, 0.f, 0.f, 0.f, 0.f};
    #pragma unroll
    for (int kk = 0; kk < 25; ++kk) {
      int k = kk * 4 + hi * 2;
      v2f b;
      b.x = wrow[k];
      b.y = wrow[k + 1];
      c = __builtin_amdgcn_wmma_f32_16x16x4_f32(false, areg[kk], false, b,
                                                (short)0, c, false, false);
    }
    #pragma unroll
    for (int g = 0; g < 8; ++g) {
      int r = wave * 16 + g + hi * 8;
      if (r < N) hx[(long)r * HD + colb + l16] = c[g];
    }
  }
}

// ---------------- per-(node,head) attention scores ----------------
__global__ void score_kernel(const float* __restrict__ hx, const float* __restrict__ al,
                             float* __restrict__ si, float* __restrict__ sj, int N) {
  int wave = (int)(((long)blockIdx.x * blockDim.x + threadIdx.x) >> 5);
  int lane = threadIdx.x & 31;
  if (wave >= N * HH) return;
  int n = wave / HH, hh = wave % HH;
  const float* base = hx + (long)n * HD + hh * PH;
  const float* ai = al + hh * 2 * PH;
  const float* aj = ai + PH;
  float s0 = 0.f, s1 = 0.f;
  for (int p = lane; p < PH; p += 32) {
    float v = base[p];
    s0 += v * ai[p];
    s1 += v * aj[p];
  }
  s0 = wave_sum(s0);
  s1 = wave_sum(s1);
  if (lane == 0) { si[n * HH + hh] = s0; sj[n * HH + hh] = s1; }
}

// ---------------- weff[h,d] = sum_p We[h*PH+p, d] * a_j[h,p]  ----------------
__global__ void weff_kernel(const float* __restrict__ Wel, const float* __restrict__ al,
                            float* __restrict__ weff) {
  int i = blockIdx.x * blockDim.x + threadIdx.x;
  if (i >= HH * DD) return;
  int hh = i / DD, d = i % DD;
  const float* aj = al + hh * 2 * PH + PH;
  float s = 0.f;
  for (int p = 0; p < PH; ++p) s += Wel[(long)(hh * PH + p) * DD + d] * aj[p];
  weff[i] = s;
}

// ---------------- escore[e,h] = ea[e,:] . weff[h,:]  (real edges only) ----------------
__global__ void escore_kernel(const float* __restrict__ ea, const float* __restrict__ weff,
                              float* __restrict__ esc, int E) {
  int wave = (int)(((long)blockIdx.x * blockDim.x + threadIdx.x) >> 5);
  int lane = threadIdx.x & 31;
  if (wave >= E) return;
  const float* row = ea + (long)wave * DD;
  float s0 = 0.f, s1 = 0.f, s2 = 0.f, s3 = 0.f;
  for (int d = lane; d < DD; d += 32) {
    float v = row[d];
    s0 += v * weff[0 * DD + d];
    s1 += v * weff[1 * DD + d];
    s2 += v * weff[2 * DD + d];
    s3 += v * weff[3 * DD + d];
  }
  s0 = wave_sum(s0); s1 = wave_sum(s1); s2 = wave_sum(s2); s3 = wave_sum(s3);
  if (lane == 0) {
    float* o = esc + (long)wave * HH;
    o[0] = s0; o[1] = s1; o[2] = s2; o[3] = s3;
  }
}

// ---------------- alpha + leaky-relu + segment max (sign-safe int trick) ----------------
__global__ void alpha_kernel(const int* __restrict__ ei, const float* __restrict__ si,
                             const float* __restrict__ sj, const float* __restrict__ esc,
                             float* __restrict__ alpha, int* __restrict__ amax,
                             int N, int E, int Etot) {
  int e = (int)((long)blockIdx.x * blockDim.x + threadIdx.x);
  if (e >= Etot) return;
  int src, dst;
  if (e < E) { src = ei[e]; dst = ei[E + e]; } else { src = dst = e - E; }
  #pragma unroll
  for (int hh = 0; hh < HH; ++hh) {
    float v = si[dst * HH + hh] + sj[src * HH + hh] + esc[(long)e * HH + hh];
    v = v > 0.f ? v : 0.2f * v;
    alpha[(long)e * HH + hh] = v;
    int* addr = amax + dst * HH + hh;
    if (v >= 0.f) atomicMax(addr, __float_as_int(v));
    else          atomicMin((unsigned int*)addr, __float_as_uint(v));
  }
}

// ---------------- exp(alpha - max) + segment sum ----------------
__global__ void expsum_kernel(const int* __restrict__ ei, float* __restrict__ alpha,
                              const int* __restrict__ amax, float* __restrict__ denom,
                              int N, int E, int Etot) {
  int e = (int)((long)blockIdx.x * blockDim.x + threadIdx.x);
  if (e >= Etot) return;
  int dst = (e < E) ? ei[E + e] : (e - E);
  #pragma unroll
  for (int hh = 0; hh < HH; ++hh) {
    float m = __int_as_float(amax[dst * HH + hh]);
    float w = __expf(alpha[(long)e * HH + hh] - m);
    alpha[(long)e * HH + hh] = w;
    atomicAdd(denom + dst * HH + hh, w);
  }
}

// ---------------- normalize + weighted head-sum of gathered hx, scatter-add ----------------
__global__ void aggr_kernel(const int* __restrict__ ei, const float* __restrict__ alpha,
                            const float* __restrict__ denom, const float* __restrict__ hx,
                            float* __restrict__ aggr, int N, int E, int Etot) {
  int wave = (int)(((long)blockIdx.x * blockDim.x + threadIdx.x) >> 5);
  int lane = threadIdx.x & 31;
  if (wave >= Etot) return;
  int src, dst;
  if (wave < E) { src = ei[wave]; dst = ei[E + wave]; } else { src = dst = wave - E; }
  float c0 = alpha[(long)wave * HH + 0] / (denom[dst * HH + 0] + 1e-16f);
  float c1 = alpha[(long)wave * HH + 1] / (denom[dst * HH + 1] + 1e-16f);
  float c2 = alpha[(long)wave * HH + 2] / (denom[dst * HH + 2] + 1e-16f);
  float c3 = alpha[(long)wave * HH + 3] / (denom[dst * HH + 3] + 1e-16f);
  const float* xb = hx + (long)src * HD;
  for (int p = lane; p < PH; p += 32) {
    float m = c0 * xb[p] + c1 * xb[PH + p] + c2 * xb[2 * PH + p] + c3 * xb[3 * PH + p];
    atomicAdd(aggr + (long)dst * DD + p, m);
  }
}

// ---------------- LayerNorm + ELU + residual (in-place on h) ----------------
__global__ void ln_kernel(const float* __restrict__ aggr, const float* __restrict__ g,
                          const float* __restrict__ b, float* __restrict__ h, int N) {
  int wave = (int)(((long)blockIdx.x * blockDim.x + threadIdx.x) >> 5);
  int lane = threadIdx.x & 31;
  if (wave >= N) return;
  const float* row = aggr + (long)wave * DD;
  float s = 0.f, ss = 0.f;
  for (int p = lane; p < DD; p += 32) {
    float v = row[p];
    s += v; ss += v * v;
  }
  s = wave_sum(s); ss = wave_sum(ss);
  float mu  = s / DD;
  float var = ss / DD - mu * mu;
  float inv = rsqrtf(var + 1e-5f);
  for (int p = lane; p < DD; p += 32) {
    float y = (row[p] - mu) * inv * g[p] + b[p];
    y = y > 0.f ? y : (__expf(y) - 1.0f);
    h[(long)wave * DD + p] += y;
  }
}

// ---------------- graph mean pooling ----------------
__global__ void pool_kernel(const float* __restrict__ h, const int* __restrict__ batch,
                            float* __restrict__ psum, float* __restrict__ pcnt, int N) {
  int wave = (int)(((long)blockIdx.x * blockDim.x + threadIdx.x) >> 5);
  int lane = threadIdx.x & 31;
  if (wave >= N) return;
  int g = batch[wave];
  for (int p = lane; p < DD; p += 32)
    atomicAdd(psum + (long)g * DD + p, h[(long)wave * DD + p]);
  if (lane == 0) atomicAdd(pcnt + g, 1.0f);
}

__global__ void pooled_kernel(const float* __restrict__ psum, const float* __restrict__ pcnt,
                              float* __restrict__ pooled, int G) {
  int i = blockIdx.x * blockDim.x + threadIdx.x;
  if (i >= G * DD) return;
  int g = i / DD;
  pooled[i] = psum[i] / fmaxf(pcnt[g], 1.0f);
}

// ---------------- dense head: hh = pooled @ w1^T + b1 ; out = hh @ w2^T + b2 ----------------
__global__ void hh_kernel(const float* __restrict__ pooled, const float* __restrict__ w1,
                          const float* __restrict__ b1, float* __restrict__ dout, int G) {
  int i = blockIdx.x * blockDim.x + threadIdx.x;
  if (i >= G * ODIM) return;
  int g = i / ODIM, o = i % ODIM;
  float s = b1[o];
  const float* pr = pooled + (long)g * DD;
  const float* wr = w1 + (long)o * DD;
  for (int d = 0; d < DD; ++d) s += pr[d] * wr[d];
  dout[G + i] = s;                     // hh stored after out (tuple order: out, hh)
}

__global__ void out_kernel(const float* __restrict__ dout_hh, const float* __restrict__ w2,
                           const float* __restrict__ b2, float* __restrict__ dout, int G) {
  int g = blockIdx.x * blockDim.x + threadIdx.x;
  if (g >= G) return;
  float s = b2[0];
  const float* hr = dout_hh + (long)g * ODIM;
  for (int o = 0; o < ODIM; ++o) s += hr[o] * w2[o];
  dout[g] = s;
}

// =====================================================================
extern "C" void kernel_launch(void* const* d_in, const int* in_sizes, int n_in,
                              void* d_out, int out_size, void* d_ws, size_t ws_size,
                              hipStream_t stream) {
  const int*   x     = (const int*)  d_in[0];
  const int*   ei    = (const int*)  d_in[1];
  const float* ea    = (const float*)d_in[2];
  const int*   batch = (const int*)  d_in[3];
  const float* emb   = (const float*)d_in[4];
  const float* W     = (const float*)d_in[5];
  const float* We    = (const float*)d_in[6];
  const float* a     = (const float*)d_in[7];
  const float* lng   = (const float*)d_in[8];
  const float* lnb   = (const float*)d_in[9];
  const float* w1    = (const float*)d_in[10];
  const float* b1    = (const float*)d_in[11];
  const float* w2    = (const float*)d_in[12];
  const float* b2    = (const float*)d_in[13];
  float* dout = (float*)d_out;

  const int N    = in_sizes[0];
  const int E    = in_sizes[1] / 2;
  const int Etot = E + N;
  const int G    = out_size / (ODIM + 1);

  // workspace layout (floats)
  float* ws = (float*)d_ws;
  long off = 0;
  float* h       = ws + off; off += (long)N * DD;
  float* hx      = ws + off; off += (long)N * HD;
  float* si      = ws + off; off += (long)N * HH;
  float* sj      = ws + off; off += (long)N * HH;
  float* weff    = ws + off; off += HH * DD;
  float* esc     = ws + off; off += (long)Etot * HH;
  float* alpha   = ws + off; off += (long)Etot * HH;
  int*   amax    = (int*)(ws + off); off += (long)N * HH;
  float* denom   = ws + off; off += (long)N * HH;
  float* aggr    = ws + off; off += (long)N * DD;
  float* psum    = ws + off; off += (long)G * DD;
  float* pcnt    = ws + off; off += G;
  float* pooled  = ws + off; off += (long)G * DD;

  const int BLK = 256;
  auto blocks  = [](long n, int b) { return (int)((n + b - 1) / b); };
  auto wblocks = [](long waves)    { return (int)((waves + 7) / 8); }; // 8 waves / 256-thr block

  // h = emb[x]
  embed_kernel<<<blocks((long)N * DD, BLK), BLK, 0, stream>>>(x, emb, h, N);

  const int ntiles = (N + 15) / 16;

  for (int l = 0; l < LY; ++l) {
    const float* Wl  = W  + (long)l * HD * DD;
    const float* Wel = We + (long)l * HD * DD;
    const float* al  = a  + (long)l * HH * 2 * PH;

    // per-layer scratch init
    fill_u32_kernel<<<blocks((long)N * HH, BLK), BLK, 0, stream>>>((unsigned*)amax, 0xFF800000u, (long)N * HH);
    fill_f32_kernel<<<blocks((long)N * HH, BLK), BLK, 0, stream>>>(denom, 0.f, (long)N * HH);
    fill_f32_kernel<<<blocks((long)N * DD, BLK), BLK, 0, stream>>>(aggr, 0.f, (long)N * DD);
    fill_f32_kernel<<<blocks((long)Etot * HH, BLK), BLK, 0, stream>>>(esc, 0.f, (long)Etot * HH);

    // hx = h @ W^T  (WMMA fp32)
    gemm_hx_kernel<<<wblocks(ntiles), BLK, 0, stream>>>(h, Wl, hx, N);
    // node scores
    score_kernel<<<wblocks((long)N * HH), BLK, 0, stream>>>(hx, al, si, sj, N);
    // folded edge-feature weights + edge scores
    weff_kernel<<<blocks(HH * DD, BLK), BLK, 0, stream>>>(Wel, al, weff);
    escore_kernel<<<wblocks(E), BLK, 0, stream>>>(ea, weff, esc, E);
    // softmax over incoming edges
    alpha_kernel<<<blocks(Etot, BLK), BLK, 0, stream>>>(ei, si, sj, esc, alpha, amax, N, E, Etot);
    expsum_kernel<<<blocks(Etot, BLK), BLK, 0, stream>>>(ei, alpha, amax, denom, N, E, Etot);
    // message aggregation
    aggr_kernel<<<wblocks(Etot), BLK, 0, stream>>>(ei, alpha, denom, hx, aggr, N, E, Etot);
    // LN + ELU + residual
    ln_kernel<<<wblocks(N), BLK, 0, stream>>>(aggr, lng + (long)l * DD, lnb + (long)l * DD, h, N);
  }

  // global mean pool + head
  fill_f32_kernel<<<blocks((long)G * DD, BLK), BLK, 0, stream>>>(psum, 0.f, (long)G * DD);
  fill_f32_kernel<<<blocks(G, BLK), BLK, 0, stream>>>(pcnt, 0.f, G);
  pool_kernel<<<wblocks(N), BLK, 0, stream>>>(h, batch, psum, pcnt, N);
  pooled_kernel<<<blocks((long)G * DD, BLK), BLK, 0, stream>>>(psum, pcnt, pooled, G);
  hh_kernel<<<blocks((long)G * ODIM, BLK), BLK, 0, stream>>>(pooled, w1, b1, dout, G);
  out_kernel<<<blocks(G, BLK), BLK, 0, stream>>>(dout + G, w2, b2, dout, G);
}